// LPOModel_50002009260264
// MI455X (gfx1250) — compile-verified
//
#include <hip/hip_runtime.h>
#include <hip/hip_bf16.h>
#include <math.h>

typedef __attribute__((ext_vector_type(2))) float v2f;
typedef __attribute__((ext_vector_type(8))) float v8f;

#define B_SZ   256
#define DZ     64
#define NS     32
#define NCOL   (B_SZ * NS)      /* 8192 (j,s) columns            */
#define NTILE  (NCOL / 16)      /* 512 column tiles per dim d    */
#define LOG_2PI 1.8378770664093453f
#define LOG_B   5.545177444479562f   /* log(256) */
#define VAR_EPS 1e-4f

// Stage 1: per-(i,d) quadratic coefficients of log p_post_i(z), stored d-major
// so the A-tile loads in the main kernel are lane-contiguous.
__global__ void lpo_coeff_kernel(const float* __restrict__ post_mean,
                                 const float* __restrict__ post_logvar,
                                 float* __restrict__ aw,
                                 float* __restrict__ bw,
                                 float* __restrict__ gw) {
    int f = blockIdx.x * blockDim.x + threadIdx.x;   // f = i*DZ + d
    if (f >= B_SZ * DZ) return;
    int i = f >> 6;
    int d = f & (DZ - 1);
    float pm  = post_mean[f];
    float plv = post_logvar[f];
    float inv = 1.0f / (2.0f * __expf(plv) + VAR_EPS);
    int o = d * B_SZ + i;
    aw[o] = -0.5f * LOG_2PI - 0.5f * plv - pm * pm * inv;  // constant term
    bw[o] = 2.0f * pm * inv;                               // linear term
    gw[o] = -inv;                                          // quadratic term
}

// Stage 2: one wave per (d, column-tile). WMMA f32 16x16x4 GEMM
// [16 i-rows x 4] x [4 x 16 (j,s)-cols] looped over 16 i-tiles, with an
// online (max-stabilized) logsumexp over i held in registers.
__global__ __launch_bounds__(128) void lpo_main_kernel(
    const float* __restrict__ prior_mean,
    const float* __restrict__ prior_logvar,
    const float* __restrict__ post_mean,
    const float* __restrict__ post_logvar,
    const float* __restrict__ eps,
    const float* __restrict__ aw,
    const float* __restrict__ bw,
    const float* __restrict__ gw,
    float* __restrict__ partials) {
    const int d    = blockIdx.y;
    const int wave = threadIdx.x >> 5;
    const int tile = blockIdx.x * 4 + wave;        // 0..NTILE-1
    const int lane = threadIdx.x & 31;
    const int n    = lane & 15;
    const bool lo  = (lane < 16);

    const int j  = tile >> 1;                      // NS/16 == 2 tiles per j
    const int s0 = (tile & 1) * 16;

    // wave-uniform posterior-j params (for the reparameterized samples)
    const int   jd  = j * DZ + d;
    const float pm  = post_mean[jd];
    const float plv = post_logvar[jd];
    const float sig = __expf(0.5f * plv);

    // per-lane sample z for column n of this tile
    const float e = eps[jd * NS + s0 + n];
    const float z = fmaf(e, sig, pm);

    // B tile (4x16 f32), rows K0..K3 = (1, z, z^2, 0):
    //   lanes 0-15 carry K0/K1, lanes 16-31 carry K2/K3 (branchless selects).
    v2f vb;
    vb.x = lo ? 1.0f : z * z;
    vb.y = lo ? z    : 0.0f;

    // online logsumexp state over i = 0..255 (this lane sees 8 rows per tile)
    float m = -INFINITY;
    float s = 0.0f;

    const float* axp  = lo ? aw : gw;              // per-lane pointer select
    const int    base = d * B_SZ;

    for (int mt = 0; mt < 16; ++mt) {
        const int i = mt * 16 + n;
        v2f va;
        va.x = axp[base + i];                      // a_i (lo) or g_i (hi)
        float bv = bw[base + i];
        va.y = lo ? bv : 0.0f;                     // b_i (lo) or 0 (hi)

        v8f c = {};
        c = __builtin_amdgcn_wmma_f32_16x16x4_f32(
                /*neg_a=*/false, va, /*neg_b=*/false, vb,
                /*c_mod=*/(short)0, c, /*reuse_a=*/false, /*reuse_b=*/false);

        // c[r] = logp_post[(lo ? r : r+8) + 16*mt][column n]
        float tmax = c[0];
        #pragma unroll
        for (int r = 1; r < 8; ++r) tmax = fmaxf(tmax, c[r]);
        float nm = fmaxf(m, tmax);
        float lsum = 0.0f;
        #pragma unroll
        for (int r = 0; r < 8; ++r) lsum += __expf(c[r] - nm);
        s = fmaf(s, __expf(m - nm), lsum);         // expf(-inf)=0 on 1st iter
        m = nm;
    }

    // merge half-wave partials: lane n has rows {0..7}+16k, lane n+16 rows {8..15}+16k
    float m2 = __shfl_xor(m, 16, 32);
    float s2 = __shfl_xor(s, 16, 32);
    float nm = fmaxf(m, m2);
    float st = s * __expf(m - nm) + s2 * __expf(m2 - nm);
    float lse = nm + __logf(st);                   // logsumexp_i logp_post

    // prior log-density at z (wave-uniform prior params for this j,d)
    const float ppm  = prior_mean[jd];
    const float pplv = prior_logvar[jd];
    const float pinv = 1.0f / (2.0f * __expf(pplv) + VAR_EPS);
    const float dzp  = z - ppm;
    const float lpp  = -0.5f * LOG_2PI - 0.5f * pplv - dzp * dzp * pinv;

    float gap = (lse - LOG_B) - lpp;
    if (!lo) gap = 0.0f;                           // hi lanes are duplicates
    #pragma unroll
    for (int off = 16; off > 0; off >>= 1)
        gap += __shfl_xor(gap, off, 32);

    if (lane == 0) partials[d * NTILE + tile] = gap;
}

// Stage 3: fixed-order deterministic reduction of all wave partials.
__global__ void lpo_reduce_kernel(const float* __restrict__ partials,
                                  float* __restrict__ out) {
    __shared__ float sm[256];
    float s = 0.0f;
    for (int t = threadIdx.x; t < DZ * NTILE; t += 256) s += partials[t];
    sm[threadIdx.x] = s;
    __syncthreads();
    for (int off = 128; off > 0; off >>= 1) {
        if ((int)threadIdx.x < off) sm[threadIdx.x] += sm[threadIdx.x + off];
        __syncthreads();
    }
    if (threadIdx.x == 0) out[0] = sm[0] * (1.0f / (float)(B_SZ * NS));
}

extern "C" void kernel_launch(void* const* d_in, const int* in_sizes, int n_in,
                              void* d_out, int out_size, void* d_ws, size_t ws_size,
                              hipStream_t stream) {
    const float* prior_mean   = (const float*)d_in[0];
    const float* prior_logvar = (const float*)d_in[1];
    const float* post_mean    = (const float*)d_in[2];
    const float* post_logvar  = (const float*)d_in[3];
    const float* eps          = (const float*)d_in[4];

    float* aw       = (float*)d_ws;        // [DZ][B_SZ]
    float* bw       = aw + B_SZ * DZ;      // [DZ][B_SZ]
    float* gw       = bw + B_SZ * DZ;      // [DZ][B_SZ]
    float* partials = gw + B_SZ * DZ;      // [DZ * NTILE] = 32768 floats

    lpo_coeff_kernel<<<dim3((B_SZ * DZ + 255) / 256), dim3(256), 0, stream>>>(
        post_mean, post_logvar, aw, bw, gw);

    lpo_main_kernel<<<dim3(NTILE / 4, DZ), dim3(128), 0, stream>>>(
        prior_mean, prior_logvar, post_mean, post_logvar, eps,
        aw, bw, gw, partials);

    lpo_reduce_kernel<<<dim3(1), dim3(256), 0, stream>>>(
        partials, (float*)d_out);
}